// AudioAttention_3659312136399
// MI455X (gfx1250) — compile-verified
//
#include <hip/hip_runtime.h>
#include <math.h>

typedef __attribute__((ext_vector_type(2))) float v2f;
typedef __attribute__((ext_vector_type(8))) float v8f;

// Fixed problem constants from the reference setup
#define NB   16      // batch B
#define NC   512     // channels C
#define NA   64      // attn dims A
#define NCA  448     // C - A
#define NG   256     // groups G
#define NHW  256     // H*W

// LDS strides (padded to avoid bank conflicts; rows stay 16B-aligned)
#define QST  68      // q_lds row stride (floats)
#define KST  68      // key slice row stride
#define SCST 260     // scores row stride
#define VST  456     // value tile row stride (456*4 B = 16B-aligned, 8-bank skew/row)

// ---------------------------------------------------------------------------
// Async global->LDS copy (CDNA5): 16B per lane, tracked by ASYNCcnt.
// GVS addressing: global address = SGPR base + per-lane VGPR byte offset; data lands at the LDS offset in vdst
// ---------------------------------------------------------------------------
__device__ __forceinline__ void async_gather_b128(unsigned lds_byte_off,
                                                  unsigned gvoff_bytes,
                                                  const float* base) {
  asm volatile("global_load_async_to_lds_b128 %0, %1, %2"
               :: "v"(lds_byte_off), "v"(gvoff_bytes), "s"(base)
               : "memory");
}
__device__ __forceinline__ void wait_async0() {
  asm volatile("s_wait_asynccnt 0x0" ::: "memory");
}

// ---------------------------------------------------------------------------
// Kernel 1: mean pool Z_img (B,C,H,W) over H*W -> z_ws[b*512 + c]
// one wave per (b,c): 32 lanes x 8 elements strided by 32, shfl-xor reduce
// ---------------------------------------------------------------------------
__global__ __launch_bounds__(256) void mean_pool_kernel(
    const float* __restrict__ Zimg, float* __restrict__ z_ws) {
  const int tid  = threadIdx.x;
  const int lane = tid & 31;
  const int bc   = blockIdx.x * 8 + (tid >> 5);   // 1024 blocks * 8 waves = 8192
  const float* src = Zimg + (size_t)bc * NHW;
  float s = 0.f;
#pragma unroll
  for (int k = 0; k < 8; ++k) s += src[lane + 32 * k];
#pragma unroll
  for (int off = 16; off >= 1; off >>= 1) s += __shfl_xor(s, off, 32);
  if (lane == 0) z_ws[bc] = s * (1.0f / (float)NHW);
}

// ---------------------------------------------------------------------------
// Kernel 2: per-group attention. 1 block (8 wave32) per group.
// ---------------------------------------------------------------------------
__global__ __launch_bounds__(256) void group_attn_kernel(
    const float* __restrict__ Zsnd,
    const int* __restrict__ pad_idx,
    const unsigned char* __restrict__ pad_mask,
    const float* __restrict__ z_ws,
    float* __restrict__ out,
    int S_max, int n_tiles) {

  // Double-buffered value tiles; key slices reuse the front of vbuf (8704 floats, spilling into buffer 1).
  __shared__ __align__(16) float vbuf[2 * 16 * VST];   // 58368 B
  __shared__ float q_lds[NB * QST];                    //  4352 B
  __shared__ float scores[NB * SCST];                  // 16640 B
  __shared__ int   idx_lds[256];                       //  1024 B
  __shared__ int   mask_lds[256];                      //  1024 B  (total 81408 B)

  const int tid  = threadIdx.x;
  const int lane = tid & 31;
  const int w    = tid >> 5;        // wave id 0..7
  const int l16  = lane & 15;
  const int hi   = lane >> 4;       // half-wave
  const int Kb   = hi * 2;          // K-pair base for fp32 A/B fragments
  const int g    = blockIdx.x;
  const int S_pad = n_tiles * 16;

  // LDS byte offset of vbuf (low 32 bits of a flat LDS pointer = LDS offset)
  const unsigned vbuf_base = (unsigned)(unsigned long long)(void*)vbuf;

  // ---- M_img: broadcast z_img (16x448) into out[g] ----
  {
    float* oimg = out + (size_t)g * (NB * NCA);
    for (int f = tid; f < NB * NCA; f += 256) {
      int b = f / NCA, c = f - b * NCA;
      oimg[f] = z_ws[b * NC + c];
    }
  }
  // ---- load Q (16x64) ----
  for (int f = tid; f < NB * NA; f += 256) {
    int b = f >> 6, a = f & 63;
    q_lds[b * QST + a] = z_ws[b * NC + NCA + a];
  }
  // ---- indices + mask ----
  for (int s = tid; s < S_pad; s += 256) {
    if (s < S_max) {
      idx_lds[s]  = pad_idx[(size_t)g * S_max + s];
      mask_lds[s] = (int)pad_mask[(size_t)g * S_max + s];
    } else {
      idx_lds[s] = 0; mask_lds[s] = 0;
    }
  }
  __syncthreads();

  // ---- scores = Q(16x64) @ K^T, one 16-col tile per wave iteration ----
  float* ks = &vbuf[w * 16 * KST];
  for (int it = 0; it < 2; ++it) {          // tiles w and w+8 (uniform barriers)
    const int t = w + 8 * it;
    const bool act = (t < n_tiles);
    if (act) {                               // gather 16 key rows (64 f32 each)
      const int r = lane >> 1, h = lane & 1;
      const int s = t * 16 + r;
      const float4* src = (const float4*)(Zsnd + (size_t)idx_lds[s] * NC + NCA + h * 32);
      float4* dst = (float4*)(ks + r * KST + h * 32);
#pragma unroll
      for (int j = 0; j < 8; ++j) dst[j] = src[j];
    }
    __syncthreads();
    if (act) {
      v8f d = {0.f, 0.f, 0.f, 0.f, 0.f, 0.f, 0.f, 0.f};
#pragma unroll
      for (int kk = 0; kk < 16; ++kk) {      // K = 64 in steps of 4
        v2f a, b;
        a.x = q_lds[l16 * QST + kk * 4 + Kb];
        a.y = q_lds[l16 * QST + kk * 4 + Kb + 1];
        b.x = ks[l16 * KST + kk * 4 + Kb];
        b.y = ks[l16 * KST + kk * 4 + Kb + 1];
        d = __builtin_amdgcn_wmma_f32_16x16x4_f32(false, a, false, b, (short)0, d,
                                                  false, false);
      }
#pragma unroll
      for (int v = 0; v < 8; ++v)
        scores[(v + 8 * hi) * SCST + t * 16 + l16] = d[v];
    }
  }
  __syncthreads();   // key reads done -> vbuf reusable; scores visible

  // ---- kick off async DMA of value tile 0 (overlaps with softmax) ----
  {
    const int st = 0, buf = 0;
#pragma unroll
    for (int j = 0; j < 7; ++j) {
      int f = tid + 256 * j;                 // 1792 float4 slots
      int r = f / 112, c4 = f - r * 112;
      unsigned ldsoff = vbuf_base + (unsigned)((buf * 16 * VST + r * VST + c4 * 4) * 4);
      unsigned goff   = (unsigned)idx_lds[st * 16 + r] * (NC * 4) + (unsigned)(c4 * 16);
      async_gather_b128(ldsoff, goff, Zsnd);
    }
  }

  // ---- softmax over S per row (wave handles rows w, w+8) ----
  for (int rr = 0; rr < 2; ++rr) {
    const int row = w + 8 * rr;
    float vals[8];
    float m = -INFINITY;
    int cnt = 0;
    for (int cc = lane; cc < S_pad; cc += 32) {
      float x = mask_lds[cc] ? scores[row * SCST + cc] : -INFINITY;
      vals[cnt++] = x;
      m = fmaxf(m, x);
    }
#pragma unroll
    for (int off = 16; off >= 1; off >>= 1) m = fmaxf(m, __shfl_xor(m, off, 32));
    float ssum = 0.f; cnt = 0;
    for (int cc = lane; cc < S_pad; cc += 32) {
      float e = expf(vals[cnt] - m);   // exp(-inf)=0 handles the mask
      vals[cnt++] = e; ssum += e;
    }
#pragma unroll
    for (int off = 16; off >= 1; off >>= 1) ssum += __shfl_xor(ssum, off, 32);
    const float inv = 1.0f / ssum;
    cnt = 0;
    for (int cc = lane; cc < S_pad; cc += 32)
      scores[row * SCST + cc] = vals[cnt++] * inv;
  }

  // ---- M_snd = attn(16xS) @ value(Sx448): double-buffered async pipeline ----
  v8f acc[4];
#pragma unroll
  for (int i = 0; i < 4; ++i) acc[i] = (v8f){0.f,0.f,0.f,0.f,0.f,0.f,0.f,0.f};

  for (int st = 0; st < n_tiles; ++st) {
    const int cur = st & 1;
    wait_async0();       // this wave's DMA for tile st complete
    __syncthreads();     // all waves' DMA complete; all reads of buf[nxt] done

    if (st + 1 < n_tiles) {          // prefetch next tile into the other buffer
      const int nxt = cur ^ 1;
#pragma unroll
      for (int j = 0; j < 7; ++j) {
        int f = tid + 256 * j;
        int r = f / 112, c4 = f - r * 112;
        unsigned ldsoff = vbuf_base + (unsigned)((nxt * 16 * VST + r * VST + c4 * 4) * 4);
        unsigned goff   = (unsigned)idx_lds[(st + 1) * 16 + r] * (NC * 4) + (unsigned)(c4 * 16);
        async_gather_b128(ldsoff, goff, Zsnd);
      }
    }

    const float* vt = &vbuf[cur * 16 * VST];
    v2f afr[4];
#pragma unroll
    for (int kk = 0; kk < 4; ++kk) {
      afr[kk].x = scores[l16 * SCST + st * 16 + kk * 4 + Kb];
      afr[kk].y = scores[l16 * SCST + st * 16 + kk * 4 + Kb + 1];
    }
#pragma unroll
    for (int i = 0; i < 4; ++i) {
      const int ct = w + 8 * i;                 // column tile (of 28)
      if (ct < 28) {
#pragma unroll
        for (int kk = 0; kk < 4; ++kk) {
          v2f b;
          b.x = vt[(kk * 4 + Kb) * VST + ct * 16 + l16];
          b.y = vt[(kk * 4 + Kb + 1) * VST + ct * 16 + l16];
          acc[i] = __builtin_amdgcn_wmma_f32_16x16x4_f32(false, afr[kk], false, b,
                                                         (short)0, acc[i], false, false);
        }
      }
    }
  }

  // ---- store M_snd ----
  float* osnd = out + (size_t)NG * NB * NCA + (size_t)g * NB * NCA;
#pragma unroll
  for (int i = 0; i < 4; ++i) {
    const int ct = w + 8 * i;
    if (ct < 28) {
#pragma unroll
      for (int v = 0; v < 8; ++v) {
        const int M = v + 8 * hi;
        osnd[M * NCA + ct * 16 + l16] = acc[i][v];
      }
    }
  }
}

// ---------------------------------------------------------------------------
extern "C" void kernel_launch(void* const* d_in, const int* in_sizes, int n_in,
                              void* d_out, int out_size, void* d_ws, size_t ws_size,
                              hipStream_t stream) {
  const float*         Zimg     = (const float*)d_in[0];
  const float*         Zsnd     = (const float*)d_in[1];
  const int*           pad_idx  = (const int*)d_in[2];
  const unsigned char* pad_mask = (const unsigned char*)d_in[3];
  float* out  = (float*)d_out;
  float* z_ws = (float*)d_ws;   // 16*512 floats

  int S_max = in_sizes[2] / NG;
  int n_tiles = (S_max + 15) / 16;
  if (n_tiles > 16) n_tiles = 16;   // reference guarantees S_max <= 256

  mean_pool_kernel<<<(NB * NC) / 8, 256, 0, stream>>>(Zimg, z_ws);
  group_attn_kernel<<<NG, 256, 0, stream>>>(Zsnd, pad_idx, pad_mask, z_ws, out,
                                            S_max, n_tiles);
}